// SpectralEncoder_6545530159343
// MI455X (gfx1250) — compile-verified
//
#include <hip/hip_runtime.h>

// Feature width for both layers (48+16 concat == 64 == HID)
#define F 64

typedef __attribute__((ext_vector_type(2))) float v2f;
typedef __attribute__((ext_vector_type(8))) float v8f;

__device__ __forceinline__ void gatomic_add(float* p, float v) {
  // relaxed agent-scope fp32 add -> global_atomic_add_f32 (no CAS loop)
  __hip_atomic_fetch_add(p, v, __ATOMIC_RELAXED, __HIP_MEMORY_SCOPE_AGENT);
}

// ---------------- elementwise helpers ----------------
__global__ void fill_kernel(float* __restrict__ p, int n, float v) {
  int i = blockIdx.x * blockDim.x + threadIdx.x;
  if (i < n) p[i] = v;
}

__global__ void negcopy_kernel(float* __restrict__ dst, const float* __restrict__ src, int n) {
  int i = blockIdx.x * blockDim.x + threadIdx.x;
  if (i < n) dst[i] = -src[i];
}

__global__ void concat_kernel(const float* __restrict__ x, const float* __restrict__ pe,
                              float* __restrict__ xc, int n) {
  int i = blockIdx.x * blockDim.x + threadIdx.x;
  if (i >= n * F) return;
  int node = i >> 6, c = i & 63;
  xc[i] = (c < 48) ? x[node * 48 + c] : pe[node * 16 + (c - 48)];
}

__global__ void bias_relu_kernel(float* __restrict__ h, const float* __restrict__ b, int total) {
  int i = blockIdx.x * blockDim.x + threadIdx.x;
  if (i >= total) return;
  float v = h[i] + b[i & 63];
  h[i] = v > 0.f ? v : 0.f;
}

// ---------------- edge normalization ----------------
__global__ void deg_kernel(const int* __restrict__ ei, float* __restrict__ deg, int nE) {
  int e = blockIdx.x * blockDim.x + threadIdx.x;
  if (e >= nE) return;
  int r = ei[e], c = ei[nE + e];
  if (r != c) gatomic_add(&deg[r], 1.0f);
}

__global__ void dis_kernel(float* __restrict__ deg, int n) {
  int i = blockIdx.x * blockDim.x + threadIdx.x;
  if (i >= n) return;
  float d = deg[i];
  deg[i] = d > 0.f ? rsqrtf(d) : 0.f;  // deg integer-valued; max(deg,1)==deg when deg>0
}

__global__ void norm_kernel(const int* __restrict__ ei, const float* __restrict__ dis,
                            float* __restrict__ nrm, int nE) {
  int e = blockIdx.x * blockDim.x + threadIdx.x;
  if (e >= nE) return;
  int r = ei[e], c = ei[nE + e];
  nrm[e] = (r == c) ? 0.f : -dis[r] * dis[c];
}

// ---------------- sparse propagation: dst[col] += scale*norm[e]*src[row] ----------------
// one thread per (edge, 4-float chunk); dst pre-initialized (0 or -T_prev)
__global__ void prop_kernel(const float* __restrict__ src, float* __restrict__ dst,
                            const int* __restrict__ ei, const float* __restrict__ nrm,
                            float scale, int nE) {
  int idx = blockIdx.x * blockDim.x + threadIdx.x;
  if (idx >= nE * 16) return;
  int e = idx >> 4;
  int c = (idx & 15) << 2;
  float w = nrm[e] * scale;
  if (w == 0.f) return;
  int r = ei[e], cl = ei[nE + e];
  float4 v = *(const float4*)(src + (size_t)r * F + c);
  float* d = dst + (size_t)cl * F + c;
  gatomic_add(d + 0, w * v.x);
  gatomic_add(d + 1, w * v.y);
  gatomic_add(d + 2, w * v.z);
  gatomic_add(d + 3, w * v.w);
}

// ---------------- dense GEMM-accumulate: O[N,64] += T[N,64] @ W[64,64] ----------------
// V_WMMA_F32_16X16X4_F32: 256-thread block = 8 waves; each wave owns one 16x16 tile.
// Block covers 32 rows x 64 cols. K loop: 64 in steps of 4 -> 16 WMMAs per wave.
__global__ void gemm_wmma_kernel(const float* __restrict__ T, const float* __restrict__ W,
                                 float* __restrict__ O, int n) {
  const int lane = threadIdx.x & 31;
  const int wave = threadIdx.x >> 5;
  const int lm   = lane & 15;       // matrix M (A) / N (B,C,D) index
  const int half = lane >> 4;       // lane half selects K pair (A/B) or row half (C/D)
  const int m0 = blockIdx.x * 32 + (wave >> 2) * 16;
  const int c0 = (wave & 3) * 16;
  if (m0 + 16 > n) return;          // wave-uniform guard (N=100000 is a multiple of 32)

  // C/D layout: VGPR r -> rows m0+r (lanes 0-15) and m0+8+r (lanes 16-31), col c0+lm
  const int mBase = m0 + half * 8;
  v8f acc;
#pragma unroll
  for (int r = 0; r < 8; ++r)
    acc[r] = O[(size_t)(mBase + r) * F + c0 + lm];

  // A 16x4: lane = M, VGPR0/1 = K {0,1} (lanes 0-15) or K {2,3} (lanes 16-31)
  const float* Arow = T + (size_t)(m0 + lm) * F + half * 2;
  // B 4x16: lane = N, VGPR0/1 = K {0,1} / {2,3} mirrored across lane halves
  const float* Bp = W + (size_t)(half * 2) * F + c0 + lm;

#pragma unroll
  for (int kk = 0; kk < F; kk += 4) {
    v2f a = *(const v2f*)(Arow + kk);
    v2f b;
    b.x = Bp[(size_t)kk * F];
    b.y = Bp[(size_t)(kk + 1) * F];
    acc = __builtin_amdgcn_wmma_f32_16x16x4_f32(
        /*neg_a=*/false, a, /*neg_b=*/false, b,
        /*c_mod=*/(short)0, acc, /*reuse_a=*/false, /*reuse_b=*/false);
  }

#pragma unroll
  for (int r = 0; r < 8; ++r)
    O[(size_t)(mBase + r) * F + c0 + lm] = acc[r];
}

// ---------------- readout ----------------
__global__ void colsum_kernel(const float* __restrict__ h, float* __restrict__ sum, int n) {
  __shared__ float s[256];
  const int col = threadIdx.x & 63;
  const int sub = threadIdx.x >> 6;  // 4 partials per column per block
  float acc = 0.f;
  for (int i = blockIdx.x * 4 + sub; i < n; i += gridDim.x * 4)
    acc += h[(size_t)i * F + col];
  s[threadIdx.x] = acc;
  __syncthreads();
  if (sub == 0)
    gatomic_add(&sum[col], s[col] + s[64 + col] + s[128 + col] + s[192 + col]);
}

__global__ void head_kernel(const float* __restrict__ csum,
                            const float* __restrict__ muW, const float* __restrict__ mub,
                            const float* __restrict__ lvW, const float* __restrict__ lvb,
                            float* __restrict__ out, float invN) {
  int t = threadIdx.x;            // 0-31 -> mu, 32-63 -> logvar
  if (t >= 64) return;
  int j = t & 31;
  const float* Wm = (t < 32) ? muW : lvW;
  const float* bm = (t < 32) ? mub : lvb;
  float acc = 0.f;
  for (int k = 0; k < F; ++k) acc += csum[k] * invN * Wm[k * 32 + j];
  out[t] = acc + bm[j];
}

// ---------------- host orchestration ----------------
extern "C" void kernel_launch(void* const* d_in, const int* in_sizes, int n_in,
                              void* d_out, int out_size, void* d_ws, size_t ws_size,
                              hipStream_t stream) {
  const float* x   = (const float*)d_in[0];
  const int*   ei  = (const int*)d_in[1];
  const float* pe  = (const float*)d_in[2];
  const float* W1  = (const float*)d_in[3];
  const float* b1  = (const float*)d_in[4];
  const float* W2  = (const float*)d_in[5];
  const float* b2  = (const float*)d_in[6];
  const float* muW = (const float*)d_in[7];
  const float* mub = (const float*)d_in[8];
  const float* lvW = (const float*)d_in[9];
  const float* lvb = (const float*)d_in[10];
  float* out = (float*)d_out;

  const int N = in_sizes[0] / 48;
  const int E = in_sizes[1] / 2;
  const size_t NF = (size_t)N * F;

  float* ws = (float*)d_ws;
  size_t off = 0;
  float* dis = ws + off;  off += (size_t)((N + 63) & ~63);
  float* nrm = ws + off;  off += (size_t)((E + 63) & ~63);
  float* B0  = ws + off;  off += NF;
  float* B1  = ws + off;  off += NF;
  float* B2  = ws + off;  off += NF;
  float* B3  = ws + off;  off += NF;
  float* csum = ws + off;

  const int TB = 256;
  auto blk = [&](size_t n) { return (int)((n + TB - 1) / TB); };
  const int gemmGrid = (N + 31) / 32;
  const int propGrid = blk((size_t)E * 16);
  const size_t Wk = (size_t)F * F;  // 4096 per Chebyshev order

  // edge normalization
  fill_kernel<<<blk(N), TB, 0, stream>>>(dis, N, 0.f);
  deg_kernel<<<blk(E), TB, 0, stream>>>(ei, dis, E);
  dis_kernel<<<blk(N), TB, 0, stream>>>(dis, N);
  norm_kernel<<<blk(E), TB, 0, stream>>>(ei, dis, nrm, E);
  concat_kernel<<<blk(NF), TB, 0, stream>>>(x, pe, B0, N);

  // ---- layer 1: T0=B0, accumulator=B3 ----
  fill_kernel<<<blk(NF), TB, 0, stream>>>(B3, (int)NF, 0.f);
  gemm_wmma_kernel<<<gemmGrid, TB, 0, stream>>>(B0, W1 + 0 * Wk, B3, N);
  fill_kernel<<<blk(NF), TB, 0, stream>>>(B1, (int)NF, 0.f);            // T1 = P(T0)
  prop_kernel<<<propGrid, TB, 0, stream>>>(B0, B1, ei, nrm, 1.f, E);
  gemm_wmma_kernel<<<gemmGrid, TB, 0, stream>>>(B1, W1 + 1 * Wk, B3, N);
  negcopy_kernel<<<blk(NF), TB, 0, stream>>>(B2, B0, (int)NF);          // T2 = 2P(T1) - T0
  prop_kernel<<<propGrid, TB, 0, stream>>>(B1, B2, ei, nrm, 2.f, E);
  gemm_wmma_kernel<<<gemmGrid, TB, 0, stream>>>(B2, W1 + 2 * Wk, B3, N);
  negcopy_kernel<<<blk(NF), TB, 0, stream>>>(B0, B1, (int)NF);          // T3 = 2P(T2) - T1
  prop_kernel<<<propGrid, TB, 0, stream>>>(B2, B0, ei, nrm, 2.f, E);
  gemm_wmma_kernel<<<gemmGrid, TB, 0, stream>>>(B0, W1 + 3 * Wk, B3, N);
  bias_relu_kernel<<<blk(NF), TB, 0, stream>>>(B3, b1, (int)NF);        // h = relu(out+b1)

  // ---- layer 2: T0=B3(h), accumulator=B1 ----
  fill_kernel<<<blk(NF), TB, 0, stream>>>(B1, (int)NF, 0.f);
  gemm_wmma_kernel<<<gemmGrid, TB, 0, stream>>>(B3, W2 + 0 * Wk, B1, N);
  fill_kernel<<<blk(NF), TB, 0, stream>>>(B2, (int)NF, 0.f);            // T1 = P(T0)
  prop_kernel<<<propGrid, TB, 0, stream>>>(B3, B2, ei, nrm, 1.f, E);
  gemm_wmma_kernel<<<gemmGrid, TB, 0, stream>>>(B2, W2 + 1 * Wk, B1, N);
  negcopy_kernel<<<blk(NF), TB, 0, stream>>>(B0, B3, (int)NF);          // T2 = 2P(T1) - T0
  prop_kernel<<<propGrid, TB, 0, stream>>>(B2, B0, ei, nrm, 2.f, E);
  gemm_wmma_kernel<<<gemmGrid, TB, 0, stream>>>(B0, W2 + 2 * Wk, B1, N);
  negcopy_kernel<<<blk(NF), TB, 0, stream>>>(B3, B2, (int)NF);          // T3 = 2P(T2) - T1
  prop_kernel<<<propGrid, TB, 0, stream>>>(B0, B3, ei, nrm, 2.f, E);
  gemm_wmma_kernel<<<gemmGrid, TB, 0, stream>>>(B3, W2 + 3 * Wk, B1, N);
  bias_relu_kernel<<<blk(NF), TB, 0, stream>>>(B1, b2, (int)NF);

  // ---- readout: mean pool + heads ----
  fill_kernel<<<1, 64, 0, stream>>>(csum, 64, 0.f);
  colsum_kernel<<<1024, TB, 0, stream>>>(B1, csum, N);
  head_kernel<<<1, 64, 0, stream>>>(csum, muW, mub, lvW, lvb, out, 1.0f / (float)N);
}